// NeRFDecoderHead_60146722013217
// MI455X (gfx1250) — compile-verified
//
#include <hip/hip_runtime.h>
#include <hip/hip_bf16.h>

// NeRF ray-march for MI455X (gfx1250, wave32).
// Wave owns 16 rays; cumsum compositing over 16-sample chunks is done as
// L(16x16 lower-triangular) @ P(16x16) via 4x v_wmma_f32_16x16x4_f32.
// Exact early-out: once min(cumsum,1) saturates, all later weights are 0,
// so saturated rays skip sampling and fully-saturated waves break the loop.

typedef __attribute__((ext_vector_type(2))) float v2f;
typedef __attribute__((ext_vector_type(8))) float v8f;

#define HAVE_WMMA_F32_SCAN __has_builtin(__builtin_amdgcn_wmma_f32_16x16x4_f32)

#define N_SAMPLES 287
#define N_CHUNKS  18          // ceil(287/16), sample 287 padded with prob=0
#define CH_STRIDE 640000      // 200*200*16
#define STEP      0.2f        // stepsize(0.5) * voxel_size(0.4)

// grid layout [C, x(200), y(200), z(16)], z fastest
__device__ __forceinline__ void tri_setup(float px, float py, float pz,
                                          int& b, float& wx, float& wy, float& wz) {
  float ux = (px + 40.f) * (199.f / 80.f);
  ux = fminf(fmaxf(ux, 0.f), 199.f);
  int ix = (int)ux; if (ix > 198) ix = 198;
  wx = ux - (float)ix;
  float uy = (py + 40.f) * (199.f / 80.f);
  uy = fminf(fmaxf(uy, 0.f), 199.f);
  int iy = (int)uy; if (iy > 198) iy = 198;
  wy = uy - (float)iy;
  float uz = (pz + 1.f) * (15.f / 6.4f);
  uz = fminf(fmaxf(uz, 0.f), 15.f);
  int iz = (int)uz; if (iz > 14) iz = 14;
  wz = uz - (float)iz;
  b = ix * 3200 + iy * 16 + iz;   // (+16: y+1, +3200: x+1, +1: z+1)
}

__device__ __forceinline__ float tri_fetch(const float* __restrict__ g, int b,
                                           float wx, float wy, float wz) {
  float a0 = g[b],        a1 = g[b + 1];
  float b0 = g[b + 16],   b1 = g[b + 17];
  float c0 = g[b + 3200], c1 = g[b + 3201];
  float d0 = g[b + 3216], d1 = g[b + 3217];
  float az = fmaf(a1 - a0, wz, a0);
  float bz = fmaf(b1 - b0, wz, b0);
  float cz = fmaf(c1 - c0, wz, c0);
  float dz = fmaf(d1 - d0, wz, d0);
  float e0 = fmaf(bz - az, wy, az);
  float e1 = fmaf(dz - cz, wy, cz);
  return fmaf(e1 - e0, wx, e0);
}

__global__ __launch_bounds__(256) void nerf_march_kernel(
    const float* __restrict__ ro_g, const float* __restrict__ rd_g,
    const float* __restrict__ dens_g, const float* __restrict__ rgb_g,
    float* __restrict__ out, int R) {
  const int lane = threadIdx.x & 31;
  const int wave = threadIdx.x >> 5;
  const int n = lane & 15;   // ray column within wave
  const int h = lane >> 4;   // half: samples [0..7] vs [8..15] of each chunk
  int ray = (blockIdx.x * 8 + wave) * 16 + n;
  const bool valid = ray < R;
  if (!valid) ray = R - 1;   // keep EXEC full for WMMA; stores are guarded

  const float ox = ro_g[ray * 3 + 0], oy = ro_g[ray * 3 + 1], oz = ro_g[ray * 3 + 2];
  const float dx = rd_g[ray * 3 + 0], dy = rd_g[ray * 3 + 1], dz = rd_g[ray * 3 + 2];

#if HAVE_WMMA_F32_SCAN
  // Lower-triangular scan matrix, split into four 16x4 column blocks.
  // A layout: lane l holds M = l%16; K-in-op = 2*(l/16) + vgpr. Global col j = 4k + 2h + v.
  v2f A0, A1, A2, A3;
  A0.x = (n >= 2 * h + 0)  ? 1.f : 0.f;  A0.y = (n >= 2 * h + 1)  ? 1.f : 0.f;
  A1.x = (n >= 2 * h + 4)  ? 1.f : 0.f;  A1.y = (n >= 2 * h + 5)  ? 1.f : 0.f;
  A2.x = (n >= 2 * h + 8)  ? 1.f : 0.f;  A2.y = (n >= 2 * h + 9)  ? 1.f : 0.f;
  A3.x = (n >= 2 * h + 12) ? 1.f : 0.f;  A3.y = (n >= 2 * h + 13) ? 1.f : 0.f;
#endif

  float carry = 0.f;                       // unclamped cumsum up to chunk start
  float dacc = 0.f, racc0 = 0.f, racc1 = 0.f, racc2 = 0.f;

  for (int chunk = 0; chunk < N_CHUNKS; ++chunk) {
    // Exact early-out: saturated rays can never contribute again.
    if (__all(carry >= 1.f)) break;        // wave-uniform -> EXEC stays full
    const bool sat = (carry >= 1.f);       // per-ray saturation

    const int sbase = chunk * 16 + h * 8;  // this lane covers samples sbase..sbase+7
    float prob[8];
    int   tb[8]; float twx[8], twy[8], twz[8];
    unsigned insideMask = 0u;
#pragma unroll
    for (int r = 0; r < 8; ++r) {
      const int s = sbase + r;
      float pr = 0.f;
      tb[r] = 0; twx[r] = 0.f; twy[r] = 0.f; twz[r] = 0.f;
      if (!sat && s < N_SAMPLES) {
        const float z = STEP * (float)s;
        const float px = fmaf(dx, z, ox);
        const float py = fmaf(dy, z, oy);
        const float pz = fmaf(dz, z, oz);
        const bool inside = (px >= -40.f) && (px <= 40.f) &&
                            (py >= -40.f) && (py <= 40.f) &&
                            (pz >= -1.f) && (pz <= 5.4f);
        if (inside) {
          int b; float wx, wy, wz;
          tri_setup(px, py, pz, b, wx, wy, wz);
          const float dv = tri_fetch(dens_g, b, wx, wy, wz);
          pr = 1.f / (1.f + __expf(-dv));                 // sigmoid
          tb[r] = b; twx[r] = wx; twy[r] = wy; twz[r] = wz;
          insideMask |= (1u << r);
        } else if (s == N_SAMPLES - 1) {
          pr = 1.f;   // last sample forced to 1 when outside bbox
        }
      }
      prob[r] = pr;
    }

    float cum[8];  // cum[r] = unclamped cumsum at sample M = r + 8h (carry included)
#if HAVE_WMMA_F32_SCAN
    // B layout: lane l needs P[4k + 2h + v][n]; source is self or xor-16 partner.
    // Pair the (B0,B2) and (B1,B3) needs so only 4 exchanges are required.
    const float t0 = h ? prob[0] : prob[2];
    const float t1 = h ? prob[1] : prob[3];
    const float t2 = h ? prob[4] : prob[6];
    const float t3 = h ? prob[5] : prob[7];
    const float s0 = __shfl_xor(t0, 16, 32);
    const float s1 = __shfl_xor(t1, 16, 32);
    const float s2 = __shfl_xor(t2, 16, 32);
    const float s3 = __shfl_xor(t3, 16, 32);
    v2f B0, B1, B2, B3;
    B0.x = h ? s0 : prob[0];  B0.y = h ? s1 : prob[1];
    B1.x = h ? s2 : prob[4];  B1.y = h ? s3 : prob[5];
    B2.x = h ? prob[2] : s0;  B2.y = h ? prob[3] : s1;
    B3.x = h ? prob[6] : s2;  B3.y = h ? prob[7] : s3;
    v8f acc;
#pragma unroll
    for (int i = 0; i < 8; ++i) acc[i] = carry;   // C = carry broadcast per ray col
    acc = __builtin_amdgcn_wmma_f32_16x16x4_f32(false, A0, false, B0, (short)0, acc, false, false);
    acc = __builtin_amdgcn_wmma_f32_16x16x4_f32(false, A1, false, B1, (short)0, acc, false, false);
    acc = __builtin_amdgcn_wmma_f32_16x16x4_f32(false, A2, false, B2, (short)0, acc, false, false);
    acc = __builtin_amdgcn_wmma_f32_16x16x4_f32(false, A3, false, B3, (short)0, acc, false, false);
#pragma unroll
    for (int r = 0; r < 8; ++r) cum[r] = acc[r];
#else
    // Fallback: per-lane serial scan + half-carry via one shuffle.
    float c = 0.f;
#pragma unroll
    for (int r = 0; r < 8; ++r) { c += prob[r]; cum[r] = c; }
    const float lo_total = __shfl_xor(cum[7], 16, 32);   // hi lanes get lo-half sum
    const float add = carry + (h ? lo_total : 0.f);
#pragma unroll
    for (int r = 0; r < 8; ++r) cum[r] += add;
#endif

    // w[m] = min(cum[m],1) - min(cum[m-1],1); hi half needs cum[M=7] from partner.
    const float cum7x = __shfl_xor(cum[7], 16, 32);
    float prevcl = fminf(h ? cum7x : carry, 1.f);
#pragma unroll
    for (int r = 0; r < 8; ++r) {
      const float cl = fminf(cum[r], 1.f);
      const float w = cl - prevcl;
      prevcl = cl;
      dacc = fmaf(w, STEP * (float)(sbase + r), dacc);
      // RGB gathers only where they can contribute (w > 0 and inside bbox).
      if ((insideMask >> r) & 1u) {
        if (w > 0.f) {
          racc0 = fmaf(w, tri_fetch(rgb_g,                 tb[r], twx[r], twy[r], twz[r]), racc0);
          racc1 = fmaf(w, tri_fetch(rgb_g +     CH_STRIDE, tb[r], twx[r], twy[r], twz[r]), racc1);
          racc2 = fmaf(w, tri_fetch(rgb_g + 2 * CH_STRIDE, tb[r], twx[r], twy[r], twz[r]), racc2);
        }
      }
    }
    carry = h ? cum[7] : cum7x;   // cum at M=15 -> new carry for both halves
  }

  // Combine the two half-lanes of each ray.
  dacc  += __shfl_xor(dacc, 16, 32);
  racc0 += __shfl_xor(racc0, 16, 32);
  racc1 += __shfl_xor(racc1, 16, 32);
  racc2 += __shfl_xor(racc2, 16, 32);

  if (valid && h == 0) {
    out[ray] = dacc;                       // depth, (R,)
    float* ro = out + R + ray * 3;         // rgb_marched, (R,3)
    ro[0] = racc0; ro[1] = racc1; ro[2] = racc2;
  }
}

extern "C" void kernel_launch(void* const* d_in, const int* in_sizes, int n_in,
                              void* d_out, int out_size, void* d_ws, size_t ws_size,
                              hipStream_t stream) {
  (void)n_in; (void)out_size; (void)d_ws; (void)ws_size;
  const float* rays_o = (const float*)d_in[0];
  const float* rays_d = (const float*)d_in[1];
  const float* voxel  = (const float*)d_in[2];
  const float* rgbrec = (const float*)d_in[3];
  float* out = (float*)d_out;
  const int R = in_sizes[0] / 3;           // 86400
  const int rays_per_block = 128;          // 8 waves x 16 rays
  const int grid = (R + rays_per_block - 1) / rays_per_block;
  nerf_march_kernel<<<grid, 256, 0, stream>>>(rays_o, rays_d, voxel, rgbrec, out, R);
}